// QuantumLayer_1082331759393
// MI455X (gfx1250) — compile-verified
//
#include <hip/hip_runtime.h>
#include <hip/hip_bf16.h>
#include <stdint.h>

// out[b][q] = prod_{j=0..q} cos(x[b][j])   (RZ phases cancel in |amp|^2,
// CNOT chain is the prefix-XOR permutation, q_weights drop out entirely).
// Memory-bound: ~80 MB traffic -> ~3.5 us @ 23.3 TB/s.

#define ROWS_PER_BLOCK 256
#define THREADS 256

typedef unsigned int uint32x4_t __attribute__((ext_vector_type(4)));
typedef int          int32x4_t  __attribute__((ext_vector_type(4)));
typedef int          int32x8_t  __attribute__((ext_vector_type(8)));

__global__ __launch_bounds__(THREADS)
void qlayer_tdm_kernel(const float* __restrict__ x,
                       float* __restrict__ out,
                       int batch) {
    __shared__ float tile[ROWS_PER_BLOCK * 4];   // 4 KB: 4 floats per row

    const uint32_t blk  = blockIdx.x;
    const uint64_t row0 = (uint64_t)blk * ROWS_PER_BLOCK;
    // rows this block actually covers (uniform)
    const uint32_t rows = (uint32_t)((row0 + ROWS_PER_BLOCK <= (uint64_t)batch)
                                     ? ROWS_PER_BLOCK
                                     : ((uint64_t)batch - row0));

    // ---- Wave 0 issues one TDM descriptor for the whole block ----
    if (threadIdx.x < 32) {
        const uint64_t gaddr = (uint64_t)(uintptr_t)(x + row0 * 16); // tile start
        // generic LDS pointer: low 32 bits are the workgroup-relative LDS byte addr
        const uint32_t lds_addr = (uint32_t)(uintptr_t)(void*)&tile[0];

        // ---- D# group 0 (128b): count=1 valid, lds_addr, global_addr, type=2 ----
        uint32x4_t g0;
        g0[0] = 1u;                                   // count=1, is_restore=0, gather off
        g0[1] = lds_addr;                             // lds_addr [63:32]
        g0[2] = (uint32_t)(gaddr & 0xFFFFFFFFu);      // global_addr[31:0]  -> [95:64]
        g0[3] = (uint32_t)((gaddr >> 32) & 0x01FFFFFFu) // global_addr[56:32] -> [120:96]
              | (2u << 30);                           // type=2 ("image")   -> [127:126]

        // ---- D# group 1 (256b) ----
        // elements are 4B; tensor viewed from tile start:
        //   tensor_dim0=16 (row length), tensor_dim1=rows, dim0_stride=16
        //   tile: 4 x rows, contiguous into LDS (no padding)
        const uint32_t tdim0   = 16u;
        const uint32_t tdim1   = rows;
        const uint32_t til0    = 4u;
        const uint32_t til1    = rows;
        const uint32_t stride0 = 16u;                 // elems between rows
        const uint32_t stride1 = 16u * ROWS_PER_BLOCK;// unused for 2D tile

        int32x8_t g1;
        g1[0] = (int)(2u << 16);                      // workgroup_mask=0, data_size=2 (4B)
        g1[1] = (int)(tdim0 << 16);                   // [63:48] tensor_dim0 lo16
        g1[2] = (int)((tdim0 >> 16) | (tdim1 << 16)); // [79:64] dim0 hi16 | [95:80] dim1 lo16
        g1[3] = (int)(((tdim1 >> 16) & 0xFFFFu) | (til0 << 16)); // dim1 hi16 | tile_dim0
        g1[4] = (int)(til1 & 0xFFFFu);                // tile_dim1 | tile_dim2=0
        g1[5] = (int)stride0;                         // tensor_dim0_stride lo32
        g1[6] = (int)((stride1 & 0xFFFFu) << 16);     // dim0_stride hi16=0 | dim1_stride lo16
        g1[7] = (int)(stride1 >> 16);                 // dim1_stride hi32

        int32x4_t z4 = {0, 0, 0, 0};                  // groups 2/3: 2D tensor, unused
        int32x8_t z8 = {0, 0, 0, 0, 0, 0, 0, 0};      // extra group (clang-23 6-arg form)

        __builtin_amdgcn_tensor_load_to_lds(g0, g1, z4, z4, z8, /*cpol=*/0);
        __builtin_amdgcn_s_wait_tensorcnt(0);
    }
    __syncthreads();   // publish LDS tile to all 8 waves

    // ---- compute: cosine prefix product ----
    const uint32_t tid = threadIdx.x;
    const uint64_t row = row0 + tid;
    if (row < (uint64_t)batch) {
        float4 v = ((const float4*)tile)[tid];
        float c0 = __cosf(v.x);
        float c1 = __cosf(v.y);
        float c2 = __cosf(v.z);
        float c3 = __cosf(v.w);
        float4 o;
        o.x = c0;
        o.y = c0 * c1;
        o.z = o.y * c2;
        o.w = o.z * c3;
        ((float4*)out)[row] = o;   // coalesced 16B store, out is (batch, 4)
    }
}

extern "C" void kernel_launch(void* const* d_in, const int* in_sizes, int n_in,
                              void* d_out, int out_size, void* d_ws, size_t ws_size,
                              hipStream_t stream) {
    const float* x = (const float*)d_in[0];
    // d_in[1] = q_weights: provably unused (RZ phases cancel in probabilities)
    (void)n_in; (void)d_ws; (void)ws_size; (void)out_size;

    const int batch  = in_sizes[0] / 16;
    const int blocks = (batch + ROWS_PER_BLOCK - 1) / ROWS_PER_BLOCK;

    qlayer_tdm_kernel<<<blocks, THREADS, 0, stream>>>(x, (float*)d_out, batch);
}